// ZBLPotential_74990128988546
// MI455X (gfx1250) — compile-verified
//
#include <hip/hip_runtime.h>
#include <hip/hip_bf16.h>

// ---------------------------------------------------------------------------
// ZBL pair potential + per-atom scatter-add (f32).
//   pair_out[p] = switched ZBL energy at r[p]
//   atom_out[i] += pair_out[p]  for i = pair_first[p]
//
// species in [1,9] => a = A0/(zi^0.23 + zj^0.23) has only 81 distinct values,
// so all r_outer-derived constants (A/3, B/4, C) live in an 81-entry LDS
// table built once per block. Hot path per pair: 1 B128 table load,
// 4 x raw v_exp_f32 (__builtin_amdgcn_exp2f; arg range |x| <= ~98 so no
// denorm fixup needed), ~15 FMAs, 1 NT store, 1 global_atomic_add_f32.
// Streaming loads/stores are non-temporal so the species table (2 MB) and
// atom accumulator (2 MB) stay resident in near caches.
// ---------------------------------------------------------------------------

#define ZBL_PREFIX 14.399645478425668f
#define ZBL_A0     0.4685f
#define ZBL_EXPOA  0.23f
#define LOG2E      1.4426950408889634f

// native clang vector types (required by __builtin_nontemporal_*)
typedef float vf4 __attribute__((ext_vector_type(4)));
typedef int   vi4 __attribute__((ext_vector_type(4)));

__global__ void zbl_zero_atoms(float* __restrict__ atom_out, int n) {
    int t = blockIdx.x * blockDim.x + threadIdx.x;
    if (t < n) atom_out[t] = 0.0f;
}

__device__ __forceinline__ float zbl_pair_val(float r, int si, int sj,
                                              const float4* __restrict__ tab) {
    // constants fold to inline immediates in the unrolled loop
    constexpr float kCoeff[4]     = {0.02817f, 0.28022f, 0.50986f, 0.18175f};
    constexpr float kNegExpoL2[4] = {-0.20162f * LOG2E, -0.4029f * LOG2E,
                                     -0.94229f * LOG2E, -3.1998f * LOG2E};

    const float zizj = ZBL_PREFIX * (float)(si * sj);
    const int idx = (si - 1) * 9 + (sj - 1);           // species guaranteed 1..9
    const float4 e = tab[idx];
    const float inv_a = e.x, A3 = e.y, B4 = e.z, C = e.w;

    // phi(r) = sum_k c_k * exp(-(EXPO_k/a) * r)  via bare hardware exp2
    float phi = 0.0f;
#pragma unroll
    for (int k = 0; k < 4; ++k) {
        phi = __fmaf_rn(kCoeff[k],
                        __builtin_amdgcn_exp2f((kNegExpoL2[k] * inv_a) * r),
                        phi);
    }
    const float Er  = __fdividef(phi, r);
    const float dr  = r - 1.0f;                        // R_INNER = 1
    const float dr2 = dr * dr;
    const float opt2 = zizj * (Er + A3 * dr * dr2 + B4 * dr2 * dr2 + C);
    const float opt1 = zizj * C;
    float v = (r < 1.0f) ? opt1 : opt2;
    return (r > 2.0f) ? 0.0f : v;                      // R_OUTER = 2
}

__launch_bounds__(256)
__global__ void zbl_pairs_kernel(const float* __restrict__ rr,
                                 const int*   __restrict__ pf,
                                 const int*   __restrict__ ps,
                                 const int*   __restrict__ species,
                                 float*       __restrict__ pair_out,
                                 float*       __restrict__ atom_out,
                                 int P) {
    __shared__ float4 tab[81];

    // ---- build 81-entry (zi,zj) table: {1/a, A/3, B/4, C} (cold path) ----
    const int t = threadIdx.x;
    if (t < 81) {
        constexpr float kCoeff[4] = {0.02817f, 0.28022f, 0.50986f, 0.18175f};
        constexpr float kExpo[4]  = {0.20162f, 0.4029f, 0.94229f, 3.1998f};
        const float zi = (float)(t / 9 + 1);
        const float zj = (float)(t % 9 + 1);
        const float a  = ZBL_A0 / (powf(zi, ZBL_EXPOA) + powf(zj, ZBL_EXPOA));
        const float inv_a = 1.0f / a;
        const float ro = 2.0f;
        float phi = 0.0f, dphi = 0.0f, d2phi = 0.0f;
#pragma unroll
        for (int k = 0; k < 4; ++k) {
            const float x  = kExpo[k] * inv_a;
            const float ee = expf(-x * ro);
            phi   += kCoeff[k] * ee;
            dphi  += kCoeff[k] * (-x) * ee;
            d2phi += kCoeff[k] * (x * x) * ee;
        }
        const float E_o   = phi / ro;
        const float dE_o  = (-phi / ro + dphi) / ro;
        const float d2E_o = (d2phi - 2.0f * dphi / ro + 2.0f * phi / (ro * ro)) / ro;
        // tc = R_OUTER - R_INNER = 1
        const float Cc = -E_o + 0.5f * dE_o - d2E_o * (1.0f / 12.0f);
        const float Bc = 2.0f * dE_o - d2E_o;
        const float Ac = -3.0f * dE_o + d2E_o;
        tab[t] = make_float4(inv_a, Ac * (1.0f / 3.0f), Bc * 0.25f, Cc);
    }
    __syncthreads();

    const int gid    = blockIdx.x * blockDim.x + t;
    const int stride = gridDim.x * blockDim.x;
    const int P4     = P >> 2;

    // ---- main: 4 pairs per thread via non-temporal 128-bit loads ----
    for (int v = gid; v < P4; v += stride) {
        const vf4 rv = __builtin_nontemporal_load(reinterpret_cast<const vf4*>(rr) + v);
        const vi4 iv = __builtin_nontemporal_load(reinterpret_cast<const vi4*>(pf) + v);
        const vi4 jv = __builtin_nontemporal_load(reinterpret_cast<const vi4*>(ps) + v);

        // hot reused table: regular (temporal) gathers
        const int si0 = species[iv.x], sj0 = species[jv.x];
        const int si1 = species[iv.y], sj1 = species[jv.y];
        const int si2 = species[iv.z], sj2 = species[jv.z];
        const int si3 = species[iv.w], sj3 = species[jv.w];

        vf4 o;
        o.x = zbl_pair_val(rv.x, si0, sj0, tab);
        o.y = zbl_pair_val(rv.y, si1, sj1, tab);
        o.z = zbl_pair_val(rv.z, si2, sj2, tab);
        o.w = zbl_pair_val(rv.w, si3, sj3, tab);

        __builtin_nontemporal_store(o, reinterpret_cast<vf4*>(pair_out) + v);

        unsafeAtomicAdd(atom_out + iv.x, o.x);
        unsafeAtomicAdd(atom_out + iv.y, o.y);
        unsafeAtomicAdd(atom_out + iv.z, o.z);
        unsafeAtomicAdd(atom_out + iv.w, o.w);
    }

    // ---- scalar tail (P not multiple of 4) ----
    for (int p = (P4 << 2) + gid; p < P; p += stride) {
        const float r = rr[p];
        const int   i = pf[p];
        const float v = zbl_pair_val(r, species[i], species[ps[p]], tab);
        pair_out[p] = v;
        unsafeAtomicAdd(atom_out + i, v);
    }
}

extern "C" void kernel_launch(void* const* d_in, const int* in_sizes, int n_in,
                              void* d_out, int out_size, void* d_ws, size_t ws_size,
                              hipStream_t stream) {
    const float* rr = (const float*)d_in[0];
    const int*   pf = (const int*)d_in[1];
    const int*   ps = (const int*)d_in[2];
    const int*   sp = (const int*)d_in[3];
    const int P = in_sizes[0];
    const int N = in_sizes[3];

    float* pair_out = (float*)d_out;
    float* atom_out = pair_out + P;

    if (N > 0) {
        zbl_zero_atoms<<<(N + 255) / 256, 256, 0, stream>>>(atom_out, N);
    }
    if (P > 0) {
        int work = (P >> 2) > 0 ? (P >> 2) : P;   // threads needed (vec4 granularity)
        int blocks = (work + 255) / 256;
        if (blocks < 1) blocks = 1;
        zbl_pairs_kernel<<<blocks, 256, 0, stream>>>(rr, pf, ps, sp, pair_out, atom_out, P);
    }
}